// ChannelFusionModule_16724602651214
// MI455X (gfx1250) — compile-verified
//
#include <hip/hip_runtime.h>
#include <hip/hip_bf16.h>
#include <math.h>

typedef float v2f __attribute__((ext_vector_type(2)));
typedef float v4f __attribute__((ext_vector_type(4)));
typedef float v8f __attribute__((ext_vector_type(8)));

// ---------------------------------------------------------------------------
// Problem constants (B=16, C=256, H=W=128)
// ---------------------------------------------------------------------------
#define BATCH 16
#define CHAN  256
#define HW    16384            // 128*128
#define TWO_C 512              // 2*C
#define C4    128              // C/4

// ---------------------------------------------------------------------------
// Kernel 1: global average pool over H,W for both tensors.
// grid.x = 2*B*C = 8192 planes, 256 threads/block, float4 coalesced loads.
// pooled layout: [B, 2C], first C entries = fft means, next C = multi means.
// ---------------------------------------------------------------------------
__global__ void cf_pool_kernel(const float* __restrict__ fft,
                               const float* __restrict__ multi,
                               float* __restrict__ pooled) {
    __shared__ float red[256];
    const int tid   = threadIdx.x;
    const int p     = blockIdx.x;          // 0..8191
    const int tensor = p >> 12;            // 0 = fft, 1 = multi
    const int pt     = p & 4095;           // plane within tensor
    const float* src = (tensor == 0) ? fft : multi;
    const v4f* src4  = (const v4f*)(src + (size_t)pt * HW);

    float s = 0.0f;
    #pragma unroll
    for (int i = 0; i < 16; ++i) {         // 16 * 256 * 4 = 16384 elements
        v4f v = src4[i * 256 + tid];
        s += v.x + v.y + v.z + v.w;
    }
    red[tid] = s;
    __syncthreads();
    #pragma unroll
    for (int off = 128; off > 0; off >>= 1) {
        if (tid < off) red[tid] += red[tid + off];
        __syncthreads();
    }
    if (tid == 0) {
        const int b = pt >> 8;             // batch
        const int c = pt & 255;            // channel
        pooled[b * TWO_C + tensor * CHAN + c] = red[0] * (1.0f / (float)HW);
    }
}

// ---------------------------------------------------------------------------
// Kernel 2: attn = sigmoid( relu(pooled @ w1^T) @ w2^T ), all via
// V_WMMA_F32_16X16X4_F32 (M=16 == batch). One block, 8 waves.
//   GEMM1: [16,512] x [512,128]  -> h   (wave w owns N-tile w)
//   GEMM2: [16,128] x [128,512]  -> attn (wave w owns N-tiles 4w..4w+3)
// A-frag (32-bit 16x4): lane m = lane&15, VGPR0/1 hold K = 2*(lane>>4)+{0,1}
// B-frag mirrors with n = lane&15.  C/D: VGPR v, lane L -> M = v + 8*(L>>4),
// N = L&15.
// ---------------------------------------------------------------------------
__global__ void cf_attn_kernel(const float* __restrict__ pooled,
                               const float* __restrict__ w1,   // [128, 512]
                               const float* __restrict__ w2,   // [512, 128]
                               float* __restrict__ attn) {     // [16, 512]
    __shared__ float h_lds[BATCH * C4];    // 16x128 = 8 KiB

    const int lane = threadIdx.x & 31;
    const int wave = threadIdx.x >> 5;     // 0..7
    const int m    = lane & 15;
    const int hi   = lane >> 4;            // 0 or 1
    const int ko   = hi * 2;               // K sub-offset {0, 2}

    // ---- GEMM1: h[b, r] = sum_c pooled[b,c] * w1[r,c], N-tile = wave ----
    v8f acc = {};
    const int n1 = wave * 16 + m;          // column (r) this lane supplies for B
    for (int kb = 0; kb < TWO_C / 4; ++kb) {
        const int k = kb * 4 + ko;
        v2f a = *(const v2f*)(pooled + m * TWO_C + k);   // A[m][k..k+1]
        v2f b = *(const v2f*)(w1 + n1 * TWO_C + k);      // B[k..k+1][n1]
        acc = __builtin_amdgcn_wmma_f32_16x16x4_f32(
            false, a, false, b, (short)0, acc, false, false);
    }
    #pragma unroll
    for (int v = 0; v < 8; ++v) {
        const int row = v + hi * 8;                      // M index
        h_lds[row * C4 + wave * 16 + m] = fmaxf(acc[v], 0.0f);
    }
    __syncthreads();

    // ---- GEMM2: attn[b, c2] = sum_r h[b,r] * w2[c2,r] ----
    for (int t = 0; t < 4; ++t) {
        const int nt = wave * 4 + t;                     // N-tile 0..31
        const int n2 = nt * 16 + m;
        v8f acc2 = {};
        for (int kb = 0; kb < C4 / 4; ++kb) {
            const int k = kb * 4 + ko;
            v2f a = *(const v2f*)(&h_lds[m * C4 + k]);   // A from LDS
            v2f b = *(const v2f*)(w2 + n2 * C4 + k);
            acc2 = __builtin_amdgcn_wmma_f32_16x16x4_f32(
                false, a, false, b, (short)0, acc2, false, false);
        }
        #pragma unroll
        for (int v = 0; v < 8; ++v) {
            const int row = v + hi * 8;
            const float x = acc2[v];
            attn[row * TWO_C + nt * 16 + m] = 1.0f / (1.0f + __expf(-x));
        }
    }
}

// ---------------------------------------------------------------------------
// Kernel 3: out = attn[:, :C] * fft + attn[:, C:] * multi   (elementwise)
// grid.x = 2*B*C*? : each block handles 1024 contiguous elements of one
// (b,c) plane -> 16 blocks per plane, 4096 planes -> 65536 blocks.
// The two weights are block-uniform; output stored non-temporally (never
// re-read) so the 192 MB L2 keeps serving the input re-reads from pass 1.
// ---------------------------------------------------------------------------
__global__ void cf_scale_kernel(const float* __restrict__ fft,
                                const float* __restrict__ multi,
                                const float* __restrict__ attn,
                                float* __restrict__ out) {
    const int blk   = blockIdx.x;
    const int plane = blk >> 4;            // 0..4095 : (b, c)
    const int sub   = blk & 15;            // 16 blocks per plane
    const int b     = plane >> 8;
    const int c     = plane & 255;

    const float wf = attn[b * TWO_C + c];
    const float wm = attn[b * TWO_C + CHAN + c];

    const size_t base = (size_t)plane * HW + (size_t)sub * 1024;
    const v4f* f4 = (const v4f*)(fft   + base);
    const v4f* m4 = (const v4f*)(multi + base);
    v4f*       o4 = (v4f*)(out + base);

    const int t = threadIdx.x;
    v4f a = f4[t];
    v4f m = m4[t];
    v4f o;
    o.x = wf * a.x + wm * m.x;
    o.y = wf * a.y + wm * m.y;
    o.z = wf * a.z + wm * m.z;
    o.w = wf * a.w + wm * m.w;
    __builtin_nontemporal_store(o, &o4[t]);
}

// ---------------------------------------------------------------------------
extern "C" void kernel_launch(void* const* d_in, const int* in_sizes, int n_in,
                              void* d_out, int out_size, void* d_ws, size_t ws_size,
                              hipStream_t stream) {
    const float* fft   = (const float*)d_in[0];
    const float* multi = (const float*)d_in[1];
    const float* w1    = (const float*)d_in[2];   // [128, 512]
    const float* w2    = (const float*)d_in[3];   // [512, 128]
    float* out = (float*)d_out;

    float* pooled = (float*)d_ws;                 // [16, 512]
    float* attn   = pooled + BATCH * TWO_C;       // [16, 512]

    cf_pool_kernel <<<2 * BATCH * CHAN, 256, 0, stream>>>(fft, multi, pooled);
    cf_attn_kernel <<<1, 256, 0, stream>>>(pooled, w1, w2, attn);
    cf_scale_kernel<<<(2 * BATCH * CHAN * HW) / (2 * 1024), 256, 0, stream>>>(
        fft, multi, attn, out);
}